// GATLayer_4054449127659
// MI455X (gfx1250) — compile-verified
//
#include <hip/hip_runtime.h>
#include <math.h>

#define D     256
#define NTOK  2048
#define BATCH 8

typedef float v2f __attribute__((ext_vector_type(2)));
typedef float v8f __attribute__((ext_vector_type(8)));

__device__ __forceinline__ float wave_reduce_add(float v) {
#pragma unroll
  for (int m = 16; m >= 1; m >>= 1) v += __shfl_xor(v, m, 32);
  return v;
}

// u[d] = sum_e Wk[d,e] * wk[e]    (one block of 256 threads per call)
__global__ void k_proj_vec(const float* __restrict__ Wk,
                           const float* __restrict__ wk,
                           float* __restrict__ u) {
  int d = threadIdx.x;
  float acc = 0.f;
  for (int e = 0; e < D; ++e) acc += Wk[d * D + e] * wk[e];
  u[d] = acc;
}

// c[idx] = X[idx,:] . u   — one wave per token, 8 waves per block
__global__ void k_dotc(const float* __restrict__ X,
                       const float* __restrict__ u,
                       float* __restrict__ c) {
  int wave = threadIdx.x >> 5, lane = threadIdx.x & 31;
  int idx = blockIdx.x * 8 + wave;
  const float* row = X + idx * D;
  float acc = 0.f;
#pragma unroll
  for (int j = 0; j < 8; ++j) {
    int d = lane + 32 * j;
    acc += row[d] * u[d];
  }
  acc = wave_reduce_add(acc);
  if (lane == 0) c[idx] = acc;
}

// per-batch masked softmax over n, then xbar[b,:] = sum_n w[b,n] * X[b,n,:]
__global__ void k_softmax_wsum(const float* __restrict__ X,
                               const float* __restrict__ c,
                               const int* __restrict__ mask,
                               float* __restrict__ xbar) {
  int b = blockIdx.x;
  int tid = threadIdx.x;                 // 1024 threads
  __shared__ float red[1024];
  const float* cb = c + b * NTOK;
  const int* mb = mask ? (mask + b * NTOK) : nullptr;

  // 1) row max over unmasked entries
  float mx = -3.0e38f;
  for (int n = tid; n < NTOK; n += 1024) {
    bool ok = (!mb) || (mb[n] != 0);
    if (ok) mx = fmaxf(mx, cb[n]);
  }
  red[tid] = mx; __syncthreads();
  for (int s = 512; s >= 1; s >>= 1) {
    if (tid < s) red[tid] = fmaxf(red[tid], red[tid + s]);
    __syncthreads();
  }
  mx = red[0]; __syncthreads();

  // 2) partition sum
  float sum = 0.f;
  for (int n = tid; n < NTOK; n += 1024) {
    bool ok = (!mb) || (mb[n] != 0);
    sum += ok ? expf(cb[n] - mx) : 0.f;
  }
  red[tid] = sum; __syncthreads();
  for (int s = 512; s >= 1; s >>= 1) {
    if (tid < s) red[tid] += red[tid + s];
    __syncthreads();
  }
  float Z = fmaxf(red[0], 1e-30f); __syncthreads();

  // 3) weighted sum: thread (d = tid&255, r = tid>>8) walks n = r, r+4, ...
  int d = tid & (D - 1);
  int r = tid >> 8;
  float acc = 0.f;
  for (int n = r; n < NTOK; n += 4) {
    bool ok = (!mb) || (mb[n] != 0);
    float w = ok ? expf(cb[n] - mx) : 0.f;
    acc += w * X[(b * NTOK + n) * D + d];
  }
  red[tid] = acc; __syncthreads();
  if (r == 0)
    xbar[b * D + d] = (red[d] + red[d + 256] + red[d + 512] + red[d + 768]) / Z;
}

// ctx[0:8, 0:256] = xbar(8x256, zero-padded to 16) @ Wv(256x256) + bv
// One block, 16 waves; wave w owns output columns [16w, 16w+16).
// Uses V_WMMA_F32_16X16X4_F32 (fp32 in, fp32 acc).
__global__ void k_ctx_wmma(const float* __restrict__ xbar,
                           const float* __restrict__ Wv,
                           const float* __restrict__ bv,
                           float* __restrict__ ctx) {
  __shared__ float As[16 * D];           // A, zero-padded rows 8..15
  int tid = threadIdx.x;                 // 512 threads
  for (int i = tid; i < 16 * D; i += 512)
    As[i] = (i < 8 * D) ? xbar[i] : 0.f;
  __syncthreads();

  int wave = tid >> 5, lane = tid & 31;
  int row = lane & 15, hi = lane >> 4;   // hi selects K+2/K+3 half
  int N = wave * 16 + row;

  v8f acc = {};
#pragma unroll 4
  for (int kk = 0; kk < D; kk += 4) {
    int ka = kk + 2 * hi;
    v2f a, bf;
    a.x  = As[row * D + ka];             // A[M=row, K=ka]
    a.y  = As[row * D + ka + 1];         // A[M=row, K=ka+1]
    bf.x = Wv[ka * D + N];               // B[K=ka,   N]
    bf.y = Wv[(ka + 1) * D + N];         // B[K=ka+1, N]
    acc = __builtin_amdgcn_wmma_f32_16x16x4_f32(
        /*neg_a=*/false, a, /*neg_b=*/false, bf,
        /*c_mod=*/(short)0, acc, /*reuse_a=*/false, /*reuse_b=*/false);
  }
  // D layout: VGPR v -> M=v (lanes 0-15), M=v+8 (lanes 16-31). Rows 8..15 are pad.
  if (hi == 0) {
#pragma unroll
    for (int v = 0; v < 8; ++v) ctx[v * D + N] = acc[v] + bv[N];
  }
}

// gconst[b] = ctx[b,:] . Wg[0:D] + bg   (one block, wave b per batch)
__global__ void k_gconst(const float* __restrict__ ctx,
                         const float* __restrict__ Wg,
                         const float* __restrict__ bg,
                         float* __restrict__ gconst) {
  int b = threadIdx.x >> 5, lane = threadIdx.x & 31;  // 256 threads
  float acc = 0.f;
#pragma unroll
  for (int j = 0; j < 8; ++j) {
    int d = lane + 32 * j;
    acc += ctx[b * D + d] * Wg[d];
  }
  acc = wave_reduce_add(acc);
  if (lane == 0) gconst[b] = acc + bg[0];
}

// layer-1 finalize: x_new = 2x + g1*ctx1, and c2 = x_new . u2 (fused)
__global__ void k_final1(const float* __restrict__ x,
                         const float* __restrict__ ctx1,
                         const float* __restrict__ Wg1,
                         const float* __restrict__ gconst1,
                         const float* __restrict__ u2,
                         float* __restrict__ xout,
                         float* __restrict__ c2) {
  int wave = threadIdx.x >> 5, lane = threadIdx.x & 31;
  int idx = blockIdx.x * 8 + wave;
  int b = idx >> 11;                     // idx / 2048
  const float* row = x + idx * D;
  float xv[8];
  float h = 0.f;
#pragma unroll
  for (int j = 0; j < 8; ++j) {
    int d = lane + 32 * j;
    xv[j] = row[d];
    h += xv[j] * (Wg1[d] + Wg1[D + d]);
  }
  h = wave_reduce_add(h);
  float g = 1.f / (1.f + expf(-(gconst1[b] + h)));
  float cacc = 0.f;
#pragma unroll
  for (int j = 0; j < 8; ++j) {
    int d = lane + 32 * j;
    float xn = 2.f * xv[j] + g * ctx1[b * D + d];
    xout[idx * D + d] = xn;
    cacc += xn * u2[d];
  }
  cacc = wave_reduce_add(cacc);
  if (lane == 0) c2[idx] = cacc;
}

// layer-2 finalize: p_new = 2p + g2*ctx2
__global__ void k_final2(const float* __restrict__ p,
                         const float* __restrict__ ctx2,
                         const float* __restrict__ Wg2,
                         const float* __restrict__ gconst2,
                         float* __restrict__ pout) {
  int wave = threadIdx.x >> 5, lane = threadIdx.x & 31;
  int idx = blockIdx.x * 8 + wave;
  int b = idx >> 11;
  const float* row = p + idx * D;
  float pv[8];
  float h = 0.f;
#pragma unroll
  for (int j = 0; j < 8; ++j) {
    int d = lane + 32 * j;
    pv[j] = row[d];
    h += pv[j] * (Wg2[d] + Wg2[D + d]);
  }
  h = wave_reduce_add(h);
  float g = 1.f / (1.f + expf(-(gconst2[b] + h)));
#pragma unroll
  for (int j = 0; j < 8; ++j) {
    int d = lane + 32 * j;
    pout[idx * D + d] = 2.f * pv[j] + g * ctx2[b * D + d];
  }
}

extern "C" void kernel_launch(void* const* d_in, const int* in_sizes, int n_in,
                              void* d_out, int out_size, void* d_ws, size_t ws_size,
                              hipStream_t stream) {
  const float* x    = (const float*)d_in[0];
  const float* p    = (const float*)d_in[1];
  const int*   mask = (const int*)d_in[2];
  // ra1: Wq(3) bq(4) Wk(5) bk(6) Wv(7) bv(8) Ws(9) bs(10) Wg(11) bg(12)
  const float* W1k = (const float*)d_in[5];
  const float* W1v = (const float*)d_in[7];
  const float* b1v = (const float*)d_in[8];
  const float* W1s = (const float*)d_in[9];    // wk1 = W1s + 256
  const float* W1g = (const float*)d_in[11];
  const float* b1g = (const float*)d_in[12];
  // ra2: 13..22
  const float* W2k = (const float*)d_in[15];
  const float* W2v = (const float*)d_in[17];
  const float* b2v = (const float*)d_in[18];
  const float* W2s = (const float*)d_in[19];   // wk2 = W2s + 256
  const float* W2g = (const float*)d_in[21];
  const float* b2g = (const float*)d_in[22];

  float* out = (float*)d_out;
  float* ws  = (float*)d_ws;
  float* u1    = ws;                    // 256
  float* u2    = u1 + D;                // 256
  float* c1    = u2 + D;                // B*N
  float* c2    = c1 + BATCH * NTOK;     // B*N
  float* xbar1 = c2 + BATCH * NTOK;     // B*D
  float* xbar2 = xbar1 + BATCH * D;     // B*D
  float* ctx1  = xbar2 + BATCH * D;     // B*D
  float* ctx2  = ctx1 + BATCH * D;      // B*D
  float* gc1   = ctx2 + BATCH * D;      // B
  float* gc2   = gc1 + BATCH;           // B

  float* xout = out;                        // x_new
  float* pout = out + BATCH * NTOK * D;     // p_new

  const int tokBlocks = (BATCH * NTOK) / 8; // wave per token, 8 waves/block

  // ---- layer 1 (p_in = x, x_in = p, no mask) ----
  k_proj_vec<<<1, 256, 0, stream>>>(W1k, W1s + D, u1);
  k_proj_vec<<<1, 256, 0, stream>>>(W2k, W2s + D, u2);
  k_dotc<<<tokBlocks, 256, 0, stream>>>(p, u1, c1);
  k_softmax_wsum<<<BATCH, 1024, 0, stream>>>(p, c1, nullptr, xbar1);
  k_ctx_wmma<<<1, 512, 0, stream>>>(xbar1, W1v, b1v, ctx1);
  k_gconst<<<1, 256, 0, stream>>>(ctx1, W1g, b1g, gc1);
  k_final1<<<tokBlocks, 256, 0, stream>>>(x, ctx1, W1g, gc1, u2, xout, c2);

  // ---- layer 2 (p_in = p, x_in = x_new, masked) ----
  k_softmax_wsum<<<BATCH, 1024, 0, stream>>>(xout, c2, mask, xbar2);
  k_ctx_wmma<<<1, 512, 0, stream>>>(xbar2, W2v, b2v, ctx2);
  k_gconst<<<1, 256, 0, stream>>>(ctx2, W2g, b2g, gc2);
  k_final2<<<tokBlocks, 256, 0, stream>>>(p, ctx2, W2g, gc2, pout);
}